// DFineMultiscaleDeformableAttention_81372450390768
// MI455X (gfx1250) — compile-verified
//
#include <hip/hip_runtime.h>
#include <hip/hip_bf16.h>
#include <math.h>

// Problem constants (from reference setup)
#define BB    16
#define QQ    300
#define CC    256
#define NHH   8
#define DHH   32
#define SUMPP 16
#define SS    8500
#define MROWS (BB * QQ)        // 4800
#define NTOT  384              // 256 offset logits + 128 attn logits
#define KDIM  256

typedef __attribute__((ext_vector_type(2))) float v2f;
typedef __attribute__((ext_vector_type(8))) float v8f;

// ---------------------------------------------------------------------------
// Kernel 1: fused logits GEMM with CDNA5 V_WMMA_F32_16X16X4_F32.
// Y[4800][384] = HS[4800][256] @ [off_kernel(256x256) | attn_kernel(256x128)]
//                + [off_bias | attn_bias]
// One wave32 per 16x16 output tile; 300 M-tiles x 24 N-tiles = 7200 waves.
// ---------------------------------------------------------------------------
__global__ __launch_bounds__(128) void gemm_logits_wmma(
    const float* __restrict__ hs,    // (4800, 256)
    const float* __restrict__ offk,  // (256, 256)  flat (C, NH*SUMP*2)
    const float* __restrict__ offb,  // (256,)      flat (NH, SUMP, 2)
    const float* __restrict__ attk,  // (256, 128)  flat (C, NH*SUMP)
    const float* __restrict__ attb,  // (128,)      flat (NH, SUMP)
    float* __restrict__ Y)           // (4800, 384)
{
    const int wave = threadIdx.x >> 5;
    const int lane = threadIdx.x & 31;
    const int gid  = blockIdx.x * 4 + wave;     // 0..7199
    const int ntile = gid % 24;
    const int mtile = gid / 24;
    const int m0 = mtile * 16;
    const int n0 = ntile * 16;
    const int half = lane >> 4;                 // 0 or 1
    const int lr   = lane & 15;

    // Wave-uniform selection of the B-matrix source (keeps EXEC all-ones).
    const float* bsrc;
    const float* bias;
    int ldb, ncol;
    if (n0 < 256) { bsrc = offk; ldb = 256; bias = offb; ncol = n0 + lr; }
    else          { bsrc = attk; ldb = 128; bias = attb; ncol = n0 - 256 + lr; }

    // A 16x4 fp32 layout: lanes 0-15 hold row M=lr with K = k0+{0,1};
    //                     lanes 16-31 hold row M=lr with K = k0+{2,3}.
    // -> per K-step each lane does one contiguous float2 load.
    const float* arow = hs + (size_t)(m0 + lr) * KDIM + 2 * half;
    // B 4x16 fp32 (transposed pattern): VGPR0 lanes0-15 = row K=k0, lanes16-31
    // row K=k0+2; VGPR1 = K=k0+1 / k0+3.
    const float* bcol = bsrc + (size_t)(2 * half) * ldb + ncol;

    v8f acc = {};
#pragma unroll 4
    for (int k = 0; k < KDIM / 4; ++k) {
        v2f a = *(const v2f*)(arow + 4 * k);
        v2f b;
        b.x = bcol[(size_t)(4 * k) * ldb];
        b.y = bcol[(size_t)(4 * k + 1) * ldb];
        acc = __builtin_amdgcn_wmma_f32_16x16x4_f32(
            /*neg_a=*/false, a, /*neg_b=*/false, b,
            /*c_mod=*/(short)0, acc, /*reuse_a=*/false, /*reuse_b=*/false);
    }

    // C/D layout: VGPR r holds M = m0 + r + 8*half, N = n0 + lr.
    const float bv = bias[ncol];
#pragma unroll
    for (int r = 0; r < 8; ++r) {
        const int m = m0 + r + 8 * half;
        Y[(size_t)m * NTOT + n0 + lr] = acc[r] + bv;
    }
}

// ---------------------------------------------------------------------------
// Kernel 2: 16-wide softmax per (b,q,h); writes attn probs (second output).
// ---------------------------------------------------------------------------
__global__ __launch_bounds__(256) void softmax_attn(
    const float* __restrict__ Y,        // (4800, 384); attn logits at col 256+
    float* __restrict__ attn_out)       // (4800, 128)
{
    const int t = blockIdx.x * 256 + threadIdx.x;   // 0..38399
    if (t >= MROWS * NHH) return;
    const int h = t & 7;
    const int row = t >> 3;
    const float* lg = Y + (size_t)row * NTOT + 256 + h * SUMPP;

    float v[SUMPP];
    float m = -3.402823466e38f;
#pragma unroll
    for (int p = 0; p < SUMPP; ++p) { v[p] = lg[p]; m = fmaxf(m, v[p]); }
    float s = 0.0f;
#pragma unroll
    for (int p = 0; p < SUMPP; ++p) { v[p] = __expf(v[p] - m); s += v[p]; }
    const float inv = 1.0f / s;
    float* o = attn_out + (size_t)row * (NHH * SUMPP) + h * SUMPP;
#pragma unroll
    for (int p = 0; p < SUMPP; ++p) o[p] = v[p] * inv;
}

// ---------------------------------------------------------------------------
// Kernel 3: deformable bilinear sampling + head aggregation.
// One 256-thread block per (b,q); thread t -> (h = t/32, d = t%32).
// Threads 0..127 stage per-(h,p) sampling loc + attn weight into LDS.
// ---------------------------------------------------------------------------
__global__ __launch_bounds__(256) void sample_agg(
    const float* __restrict__ Y,     // logits (offsets incl. off_bias in cols 0..255)
    const float* __restrict__ ehs,   // (B, S, C) value tensor
    const float* __restrict__ rp,    // (B, Q, 1, 4) reference points
    const float* __restrict__ nps,   // (16,) num_points_scale
    const float* __restrict__ attn,  // (4800, 128) softmax probs
    float* __restrict__ out)         // (4800, 256)
{
    __shared__ float s_lx[NHH * SUMPP];
    __shared__ float s_ly[NHH * SUMPP];
    __shared__ float s_aw[NHH * SUMPP];

    const int row = blockIdx.x;          // b*Q + q
    const int b   = row / QQ;
    const int t   = threadIdx.x;

    if (t < NHH * SUMPP) {
        const int h = t >> 4, p = t & 15;
        const float cx = rp[row * 4 + 0];
        const float cy = rp[row * 4 + 1];
        const float ww = rp[row * 4 + 2];
        const float wh = rp[row * 4 + 3];
        const float ox = Y[(size_t)row * NTOT + h * (SUMPP * 2) + p * 2 + 0];
        const float oy = Y[(size_t)row * NTOT + h * (SUMPP * 2) + p * 2 + 1];
        const float sc = nps[p] * 0.5f;      // num_points_scale * OFFSET_SCALE
        s_lx[t] = cx + ox * sc * ww;
        s_ly[t] = cy + oy * sc * wh;
        s_aw[t] = attn[(size_t)row * (NHH * SUMPP) + t];
    }
    __syncthreads();

    const int h = t >> 5;
    const int d = t & 31;
    const int cHt[4] = {80, 40, 20, 10};
    const int cWt[4] = {80, 40, 20, 10};
    const int cSt[4] = {0, 6400, 8000, 8400};
    const int dxs[4] = {0, 1, 0, 1};
    const int dys[4] = {0, 0, 1, 1};

    // value[b, h, s, d] = ehs[b, s, h*DH + d]
    const float* vbase = ehs + (size_t)b * SS * CC + h * DHH + d;

    float acc = 0.0f;
#pragma unroll
    for (int p = 0; p < SUMPP; ++p) {
        const int lvl = p >> 2;
        const int Hh = cHt[lvl], Ww = cWt[lvl], st = cSt[lvl];
        const float lx = s_lx[h * SUMPP + p];
        const float ly = s_ly[h * SUMPP + p];
        const float aw = s_aw[h * SUMPP + p];

        const float x = lx * (float)Ww - 0.5f;
        const float y = ly * (float)Hh - 0.5f;
        const float x0f = floorf(x), y0f = floorf(y);
        const float fx = x - x0f, fy = y - y0f;
        const int x0 = (int)x0f, y0 = (int)y0f;

        const float cw[4] = {(1.0f - fx) * (1.0f - fy), fx * (1.0f - fy),
                             (1.0f - fx) * fy,          fx * fy};
#pragma unroll
        for (int c = 0; c < 4; ++c) {
            const int xc = x0 + dxs[c];
            const int yc = y0 + dys[c];
            const bool valid = (xc >= 0) & (xc < Ww) & (yc >= 0) & (yc < Hh);
            const int xi = xc < 0 ? 0 : (xc > Ww - 1 ? Ww - 1 : xc);
            const int yi = yc < 0 ? 0 : (yc > Hh - 1 ? Hh - 1 : yc);
            const float g = vbase[(size_t)(st + yi * Ww + xi) * CC];
            acc = fmaf(valid ? (cw[c] * aw) : 0.0f, g, acc);
        }
    }
    out[(size_t)row * CC + h * DHH + d] = acc;
}

// ---------------------------------------------------------------------------
extern "C" void kernel_launch(void* const* d_in, const int* in_sizes, int n_in,
                              void* d_out, int out_size, void* d_ws, size_t ws_size,
                              hipStream_t stream) {
    const float* hs   = (const float*)d_in[0];  // hidden_states (16,300,256)
    const float* ehs  = (const float*)d_in[1];  // encoder_hidden_states (16,8500,256)
    const float* rp   = (const float*)d_in[2];  // reference_points (16,300,1,4)
    const float* offk = (const float*)d_in[3];  // off_kernel (256,8,16,2)
    const float* offb = (const float*)d_in[4];  // off_bias (8,16,2)
    const float* attk = (const float*)d_in[5];  // attn_kernel (256,8,16)
    const float* attb = (const float*)d_in[6];  // attn_bias (8,16)
    const float* nps  = (const float*)d_in[7];  // num_points_scale (16,)
    // d_in[8]: spatial_shapes (constexpr in kernels)

    float* out  = (float*)d_out;                       // (4800, 256)
    float* attn = out + (size_t)MROWS * CC;            // (4800, 128)
    float* Y    = (float*)d_ws;                        // (4800, 384) logits

    // 7200 tiles, 4 waves/block -> 1800 blocks
    gemm_logits_wmma<<<1800, 128, 0, stream>>>(hs, offk, offb, attk, attb, Y);
    softmax_attn<<<(MROWS * NHH + 255) / 256, 256, 0, stream>>>(Y, attn);
    sample_agg<<<MROWS, 256, 0, stream>>>(Y, ehs, rp, nps, attn, out);
}